// SemiDiscreteOptimalTransport_48215302865480
// MI455X (gfx1250) — compile-verified
//
#include <hip/hip_runtime.h>

typedef __attribute__((ext_vector_type(2))) float v2f;
typedef __attribute__((ext_vector_type(8))) float v8f;

#define WAVES_PER_BLOCK 8
#define ROWS_PER_BLOCK 32   // two 16-row WMMA slabs per block
#define DDIM 64
#define KSTEPS 16           // 64 / 4 (K=4 per fp32 WMMA)

__global__ void psi_mean_kernel(const float* __restrict__ psi, int M,
                                float* __restrict__ mean_out) {
    __shared__ float red[256];
    float s = 0.0f;
    for (int i = threadIdx.x; i < M; i += 256) s += psi[i];
    red[threadIdx.x] = s;
    __syncthreads();
    for (int off = 128; off > 0; off >>= 1) {
        if ((int)threadIdx.x < off) red[threadIdx.x] += red[threadIdx.x + off];
        __syncthreads();
    }
    if (threadIdx.x == 0) mean_out[0] = red[0] / (float)M;
}

__global__ __launch_bounds__(256)
void sdot_min_kernel(const float* __restrict__ src,
                     const float* __restrict__ tgt,
                     const float* __restrict__ psi,
                     const float* __restrict__ meanp,
                     float* __restrict__ out,
                     int M) {
    const int tid  = threadIdx.x;
    const int lane = tid & 31;
    const int wave = tid >> 5;
    const int h    = lane >> 4;     // half-wave select (K pair / row half)
    const int l16  = lane & 15;
    const int n0   = blockIdx.x * ROWS_PER_BLOCK;

    __shared__ float lds_sq[ROWS_PER_BLOCK];
    __shared__ float lds_min[WAVES_PER_BLOCK][ROWS_PER_BLOCK];

    // ---- A fragments for both 16-row slabs; ||x||^2 folded in for free.
    // fp32 16x4 A layout: lanes 0-15 -> rows, VGPR0/1 = K pair, lanes 16-31 -> K+2.
    v2f a0[KSTEPS], a1[KSTEPS];
    float asq0 = 0.0f, asq1 = 0.0f;
    const float* s0 = src + (size_t)(n0 + l16) * DDIM;
    const float* s1 = src + (size_t)(n0 + 16 + l16) * DDIM;
#pragma unroll
    for (int kk = 0; kk < KSTEPS; ++kk) {
        const int d = 4 * kk + 2 * h;
        a0[kk] = *(const v2f*)(s0 + d);
        a1[kk] = *(const v2f*)(s1 + d);
        asq0 += a0[kk].x * a0[kk].x + a0[kk].y * a0[kk].y;
        asq1 += a1[kk].x * a1[kk].x + a1[kk].y * a1[kk].y;
    }
    // lane and its xor-16 partner each hold half of the same row's sum of squares
    asq0 += __shfl_xor(asq0, 16, 32);
    asq1 += __shfl_xor(asq1, 16, 32);
    if (wave == 0 && lane < 16) {
        lds_sq[lane]      = asq0;
        lds_sq[16 + lane] = asq1;
    }
    __syncthreads();

    // C/D layout: lane L holds rows v + 8*(L>>4), col = L&15
    float rsq0[8], rsq1[8];
#pragma unroll
    for (int v = 0; v < 8; ++v) {
        rsq0[v] = lds_sq[v + 8 * h];
        rsq1[v] = lds_sq[16 + v + 8 * h];
    }

    const float INF = __builtin_inff();
    v8f min0, min1;
#pragma unroll
    for (int v = 0; v < 8; ++v) { min0[v] = INF; min1[v] = INF; }

    // ---- Sweep this wave's M chunk, 16 targets per WMMA tile.
    const int chunk = M / WAVES_PER_BLOCK;
    const int mbeg  = wave * chunk;
    const int mend  = mbeg + chunk;
    for (int m0 = mbeg; m0 < mend; m0 += 16) {
        const float* t = tgt + (size_t)(m0 + l16) * DDIM;

        // branch-free prefetch two tiles ahead (uniform clamped index: no EXEC churn)
        int pfm = m0 + 32;
        pfm = (pfm < mend) ? pfm : mbeg;
        __builtin_prefetch(tgt + (size_t)(pfm + l16) * DDIM + 2 * h, 0, 3);

        v2f bf[KSTEPS];
        float bsq = 0.0f;
#pragma unroll
        for (int kk = 0; kk < KSTEPS; ++kk) {
            bf[kk] = *(const v2f*)(t + 4 * kk + 2 * h);
            bsq += bf[kk].x * bf[kk].x + bf[kk].y * bf[kk].y;
        }
        bsq += __shfl_xor(bsq, 16, 32);                 // full ||y_m||^2, m = m0 + l16
        const float colterm = bsq - psi[m0 + l16];      // (||y||^2 - psi) for this column

        v8f acc0 = {};
        v8f acc1 = {};
#pragma unroll
        for (int kk = 0; kk < KSTEPS; ++kk) {
            acc0 = __builtin_amdgcn_wmma_f32_16x16x4_f32(
                false, a0[kk], false, bf[kk], (short)0, acc0, false, false);
            acc1 = __builtin_amdgcn_wmma_f32_16x16x4_f32(
                false, a1[kk], false, bf[kk], (short)0, acc1, false, false);
        }

#pragma unroll
        for (int v = 0; v < 8; ++v) {
            const float c0 = rsq0[v] + colterm - 2.0f * acc0[v];
            const float c1 = rsq1[v] + colterm - 2.0f * acc1[v];
            min0[v] = fminf(min0[v], c0);
            min1[v] = fminf(min1[v], c1);
        }
    }

    // ---- Reduce min across the 16 columns held within each half-wave.
#pragma unroll
    for (int off = 1; off < 16; off <<= 1) {
#pragma unroll
        for (int v = 0; v < 8; ++v) {
            min0[v] = fminf(min0[v], __shfl_xor(min0[v], off, 32));
            min1[v] = fminf(min1[v], __shfl_xor(min1[v], off, 32));
        }
    }
    if (l16 == 0) {
#pragma unroll
        for (int v = 0; v < 8; ++v) {
            lds_min[wave][8 * h + v]      = min0[v];   // rows n0 + 8h + v
            lds_min[wave][16 + 8 * h + v] = min1[v];   // rows n0 + 16 + 8h + v
        }
    }
    __syncthreads();

    // ---- Combine across the block's 8 waves and emit.
    if (tid < ROWS_PER_BLOCK) {
        float r = lds_min[0][tid];
#pragma unroll
        for (int w = 1; w < WAVES_PER_BLOCK; ++w) r = fminf(r, lds_min[w][tid]);
        out[n0 + tid] = r + meanp[0];
    }
}

extern "C" void kernel_launch(void* const* d_in, const int* in_sizes, int n_in,
                              void* d_out, int out_size, void* d_ws, size_t ws_size,
                              hipStream_t stream) {
    const float* src = (const float*)d_in[0];   // [N, 64]
    const float* tgt = (const float*)d_in[1];   // [M, 64]
    const float* psi = (const float*)d_in[2];   // [M]
    float* out   = (float*)d_out;               // [N]
    float* meanp = (float*)d_ws;                // 1 float scratch

    const int M = in_sizes[2];
    const int N = in_sizes[0] / DDIM;

    psi_mean_kernel<<<1, 256, 0, stream>>>(psi, M, meanp);

    const int blocks = N / ROWS_PER_BLOCK;      // 8192 / 32 = 256
    sdot_min_kernel<<<blocks, 256, 0, stream>>>(src, tgt, psi, meanp, out, M);
}